// DropBlock_979252543593
// MI455X (gfx1250) — compile-verified
//
#include <hip/hip_runtime.h>
#include <stdint.h>

// DropBlock on MI455X (gfx1250): bandwidth-bound, two-pass, B128 everywhere.
// Pass 1: per-plane async-stage u into LDS (GLOBAL_LOAD_ASYNC_TO_LDS_B128,
//         ASYNCcnt), bit-parallel 5x5 max-dilation in 64-bit row words,
//         exact integer kept-count via atomics, 8 MiB bitmask spill to d_ws.
// Pass 2: out = x * (bit ? 0 : countM/count) with float4 loads/stores,
//         reading the 8 MiB bitmask instead of re-reading 225 MB of u.

#define MH 60                 // mask rows (H - bs + 1)
#define MW 60                 // mask cols
#define OH 64                 // output rows
#define PLANE_U (MH * MW)     // 3600 floats = 14400 B (16B multiple)
#define PLANE_X (OH * 64)     // 4096
#define NPLANES (64 * 256)    // 16384
#define COUNT_M 67108864.0f   // 64*256*64*64

// ---- CDNA5 async global->LDS copy (ASYNCcnt path, CDNA5 ISA) ----
__device__ __forceinline__ void async_cp16(uint32_t lds_off, uint64_t gaddr) {
  asm volatile("global_load_async_to_lds_b128 %0, %1, off"
               :: "v"(lds_off), "v"(gaddr) : "memory");
}
__device__ __forceinline__ void wait_async0() {
  asm volatile("s_wait_asynccnt 0" ::: "memory");
}

// Compute dilated 64-bit row words for one plane into `dil` (shared).
// Ends with __syncthreads(); dil[] valid for all threads afterwards.
__device__ __forceinline__ void compute_dilated(const float* __restrict__ u,
                                                float gamma,
                                                float* su,
                                                unsigned long long* rowor,
                                                unsigned long long* dil,
                                                int plane) {
  const float* up = u + (size_t)plane * PLANE_U;
  // flat LDS pointer low 32 bits == LDS byte address (ISA §10.2)
  uint32_t lbase = (uint32_t)(uintptr_t)(void*)su;
  // 14400 B = 900 x 16B chunks, 16B per lane, coalesced, async engine writes LDS
  for (int i = threadIdx.x; i < PLANE_U / 4; i += blockDim.x) {
    async_cp16(lbase + (uint32_t)i * 16u, (uint64_t)(uintptr_t)(up + i * 4));
  }
  wait_async0();
  __syncthreads();

  // Row bitmask + W-dilation: bit w of rowor = OR_{j=0..4} mask[w-j]
  // Row stride = 240 B -> float4 (ds_load_b128) legal and conflict-friendly.
  if (threadIdx.x < MH) {
    const float4* row = (const float4*)(su + threadIdx.x * MW);
    unsigned long long m = 0ull;
#pragma unroll
    for (int c = 0; c < MW / 4; ++c) {
      float4 v = row[c];
      unsigned b = (v.x < gamma ? 1u : 0u) | (v.y < gamma ? 2u : 0u) |
                   (v.z < gamma ? 4u : 0u) | (v.w < gamma ? 8u : 0u);
      m |= (unsigned long long)b << (c * 4);
    }
    m = m | (m << 1) | (m << 2) | (m << 3) | (m << 4);
    rowor[threadIdx.x] = m;
  }
  __syncthreads();

  // H-dilation: output row h covers mask rows [h-4, h] ∩ [0, 59]
  if (threadIdx.x < OH) {
    int h  = (int)threadIdx.x;
    int lo = (h - 4 < 0) ? 0 : h - 4;
    int hi = (h < MH - 1) ? h : MH - 1;
    unsigned long long d = 0ull;
    for (int r = lo; r <= hi; ++r) d |= rowor[r];
    dil[h] = d;
  }
  __syncthreads();
}

// Vectorized apply: 4 consecutive floats per lane; a 4-aligned column group
// never straddles a 64-wide row, so one 4-bit mask slice covers the float4.
__device__ __forceinline__ void apply_plane(const float* __restrict__ x,
                                            float* __restrict__ out,
                                            const unsigned long long* dil,
                                            float scale, int plane) {
  size_t base = (size_t)plane * PLANE_X;
  const float4* x4 = (const float4*)(x + base);
  float4* o4 = (float4*)(out + base);
#pragma unroll
  for (int k = 0; k < (PLANE_X / 4) / 256; ++k) {
    int i = (int)threadIdx.x + k * 256;          // float4 index in plane
    int h  = i >> 4;                             // 16 float4 per row
    int w0 = (i & 15) << 2;                      // starting column bit
    unsigned bits = (unsigned)((dil[h] >> w0) & 0xFull);
    float4 v = x4[i];
    float4 r;
    r.x = (bits & 1u) ? 0.0f : v.x * scale;
    r.y = (bits & 2u) ? 0.0f : v.y * scale;
    r.z = (bits & 4u) ? 0.0f : v.z * scale;
    r.w = (bits & 8u) ? 0.0f : v.w * scale;
    o4[i] = r;
  }
}

__global__ void __launch_bounds__(256)
dropblock_count_kernel(const float* __restrict__ u,
                       const float* __restrict__ gamma_p,
                       unsigned int* __restrict__ count,
                       unsigned long long* __restrict__ mask_out,
                       int store_mask) {
  __shared__ __align__(16) float su[PLANE_U];
  __shared__ unsigned long long rowor[MH];
  __shared__ unsigned long long dil[OH];
  __shared__ unsigned int cnt;
  if (threadIdx.x == 0) cnt = 0;     // barrier inside compute_dilated orders this
  float gamma = *gamma_p;
  compute_dilated(u, gamma, su, rowor, dil, (int)blockIdx.x);
  if (threadIdx.x < OH) {
    unsigned long long d = dil[threadIdx.x];
    atomicAdd(&cnt, 64u - (unsigned)__popcll(d));   // kept = ~dilated, all 64 bits valid
    if (store_mask)
      mask_out[(size_t)blockIdx.x * OH + threadIdx.x] = d;
  }
  __syncthreads();
  if (threadIdx.x == 0) atomicAdd(count, cnt);      // exact integer, deterministic
}

__global__ void __launch_bounds__(256)
dropblock_apply_mask_kernel(const float* __restrict__ x,
                            const unsigned long long* __restrict__ mask,
                            const unsigned int* __restrict__ count,
                            float* __restrict__ out) {
  __shared__ unsigned long long dil[OH];
  if (threadIdx.x < OH)
    dil[threadIdx.x] = mask[(size_t)blockIdx.x * OH + threadIdx.x];
  __syncthreads();
  float scale = COUNT_M / (float)(*count);
  apply_plane(x, out, dil, scale, (int)blockIdx.x);
}

__global__ void __launch_bounds__(256)
dropblock_apply_recompute_kernel(const float* __restrict__ x,
                                 const float* __restrict__ u,
                                 const float* __restrict__ gamma_p,
                                 const unsigned int* __restrict__ count,
                                 float* __restrict__ out) {
  __shared__ __align__(16) float su[PLANE_U];
  __shared__ unsigned long long rowor[MH];
  __shared__ unsigned long long dil[OH];
  float gamma = *gamma_p;
  compute_dilated(u, gamma, su, rowor, dil, (int)blockIdx.x);
  float scale = COUNT_M / (float)(*count);
  apply_plane(x, out, dil, scale, (int)blockIdx.x);
}

extern "C" void kernel_launch(void* const* d_in, const int* in_sizes, int n_in,
                              void* d_out, int out_size, void* d_ws, size_t ws_size,
                              hipStream_t stream) {
  (void)in_sizes; (void)n_in; (void)out_size;
  const float* x     = (const float*)d_in[0];
  const float* u     = (const float*)d_in[1];
  const float* gamma = (const float*)d_in[2];
  float* out = (float*)d_out;

  unsigned int* count = (unsigned int*)d_ws;
  const size_t mask_bytes = (size_t)NPLANES * OH * sizeof(unsigned long long); // 8 MiB
  const bool store = (ws_size >= 256 + mask_bytes);   // ws_size is fixed -> deterministic
  unsigned long long* maskbuf = (unsigned long long*)((char*)d_ws + 256);

  hipMemsetAsync(d_ws, 0, sizeof(unsigned int), stream);  // zero counter (capturable)

  dropblock_count_kernel<<<NPLANES, 256, 0, stream>>>(u, gamma, count, maskbuf,
                                                      store ? 1 : 0);
  if (store) {
    dropblock_apply_mask_kernel<<<NPLANES, 256, 0, stream>>>(x, maskbuf, count, out);
  } else {
    dropblock_apply_recompute_kernel<<<NPLANES, 256, 0, stream>>>(x, u, gamma, count, out);
  }
}